// GMN_FEAT_Net_3582002725359
// MI455X (gfx1250) — compile-verified
//
#include <hip/hip_runtime.h>
#include <hip/hip_bf16.h>
#include <climits>

// ------------------------------------------------------------------
// Problem constants
// ------------------------------------------------------------------
namespace {
constexpr int BB  = 16;
constexpr int N1C = 48;
constexpr int N2C = 48;
constexpr int NNC = N1C * N2C;       // 2304
constexpr int EC  = 64;
constexpr int PDC = 16;
constexpr int PHC = 32;
constexpr int DC  = 128;
constexpr float TAUC = 0.05f;

// d_out layout: ss [B*NN], x [B*NN], K [B*NN*NN]
constexpr long SS_OFF  = 0;
constexpr long X_OFF   = (long)BB * NNC;
constexpr long K_OFF   = 2L * BB * NNC;
constexpr long K_ELEMS = (long)BB * NNC * NNC;

// workspace: float region offsets (in floats)
constexpr long WS_U_SRC = 0;
constexpr long WS_F_SRC = WS_U_SRC + (long)BB * N1C * DC;
constexpr long WS_U_TGT = WS_F_SRC + (long)BB * N1C * DC;
constexpr long WS_F_TGT = WS_U_TGT + (long)BB * N2C * DC;
constexpr long WS_X     = WS_F_TGT + (long)BB * N2C * DC;
constexpr long WS_Y     = WS_X + (long)BB * 2 * DC * EC;
constexpr long WS_CO    = WS_Y + (long)BB * 2 * DC * EC;
constexpr long WS_Z     = WS_CO + 4L * DC * DC;
constexpr long WS_KE    = WS_Z + (long)BB * 2 * DC * EC;
constexpr long WS_KP    = WS_KE + (long)BB * EC * EC;
constexpr long WS_EVAL  = WS_KP + (long)BB * N1C * N2C;
constexpr long WS_X1    = WS_EVAL + 4L * BB * EC;
constexpr long WS_X2A   = WS_X1 + (long)BB * NNC * 16;
constexpr long WS_X2B   = WS_X2A + (long)BB * NNC * 16;
constexpr long WS_X6A   = WS_X2B + (long)BB * NNC * 16;
constexpr long WS_X6B   = WS_X6A + (long)BB * NNC;
constexpr long WS_FEND  = WS_X6B + (long)BB * NNC;

// int region offsets (in ints, relative to wsI = (int*)(wsF + WS_FEND))
constexpr long WI_EIDX = 0;
constexpr long WI_CROW = WI_EIDX + 4L * BB * EC;
constexpr long WI_CCOL = WI_CROW + (long)BB * EC * EC;
constexpr long WI_OWN  = WI_CCOL + (long)BB * EC * EC;
constexpr long WI_DEG  = WI_OWN + (long)BB * EC * EC;
} // namespace

typedef __attribute__((ext_vector_type(2))) float v2f_t;
typedef __attribute__((ext_vector_type(8))) float v8f_t;

// ------------------------------------------------------------------
// Node/edge MLPs: pos(16) -> 32 -> 32 -> 128, relu on every layer.
// variant: 0=U_src(node), 1=F_src(edge), 2=U_tgt(node), 3=F_tgt(edge)
// ------------------------------------------------------------------
__global__ void mlp_kernel(const float* __restrict__ pos_src, const float* __restrict__ pos_tgt,
    const float* nW0, const float* nb0, const float* nW1, const float* nb1,
    const float* nW2, const float* nb2,
    const float* eW0, const float* eb0, const float* eW1, const float* eb1,
    const float* eW2, const float* eb2,
    float* U_src, float* F_src, float* U_tgt, float* F_tgt)
{
  int idx = blockIdx.x * blockDim.x + threadIdx.x;
  if (idx >= 4 * BB * N1C) return;
  int variant = idx / (BB * N1C);
  int node = idx % (BB * N1C);
  const float* pos = (variant < 2) ? pos_src : pos_tgt;
  bool edge = (variant & 1);
  const float* W0 = edge ? eW0 : nW0; const float* b0 = edge ? eb0 : nb0;
  const float* W1 = edge ? eW1 : nW1; const float* b1 = edge ? eb1 : nb1;
  const float* W2 = edge ? eW2 : nW2; const float* b2 = edge ? eb2 : nb2;

  float in[PDC];
#pragma unroll
  for (int d = 0; d < PDC; ++d) in[d] = pos[(long)node * PDC + d];

  float h1[PHC];
  for (int o = 0; o < PHC; ++o) {
    float s = b0[o];
#pragma unroll
    for (int i2 = 0; i2 < PDC; ++i2) s += in[i2] * W0[i2 * PHC + o];
    h1[o] = fmaxf(s, 0.f);
  }
  float h2[PHC];
  for (int o = 0; o < PHC; ++o) {
    float s = b1[o];
#pragma unroll
    for (int i2 = 0; i2 < PHC; ++i2) s += h1[i2] * W1[i2 * PHC + o];
    h2[o] = fmaxf(s, 0.f);
  }
  float* out = (variant == 0) ? U_src : (variant == 1) ? F_src
             : (variant == 2) ? U_tgt : F_tgt;
  for (int o = 0; o < DC; ++o) {
    float s = b2[o];
#pragma unroll
    for (int i2 = 0; i2 < PHC; ++i2) s += h2[i2] * W2[i2 * DC + o];
    out[(long)node * DC + o] = fmaxf(s, 0.f);
  }
}

// ------------------------------------------------------------------
// X[b,c,e] = sum_n F[b,n,c&127] * (c<128 ? G : H)[b,n,e], likewise Y.
// ------------------------------------------------------------------
__global__ void xy_kernel(const float* __restrict__ F_src, const float* __restrict__ G_src,
                          const float* __restrict__ H_src,
                          const float* __restrict__ F_tgt, const float* __restrict__ G_tgt,
                          const float* __restrict__ H_tgt,
                          float* __restrict__ Xo, float* __restrict__ Yo)
{
  long idx = (long)blockIdx.x * blockDim.x + threadIdx.x;
  const long per = (long)BB * 2 * DC * EC;
  if (idx >= 2 * per) return;
  int sel = (int)(idx / per);
  long rem = idx % per;
  int b = (int)(rem / (2 * DC * EC));
  int ce = (int)(rem % (2 * DC * EC));
  int c = ce / EC, e = ce % EC;
  const float* F  = sel ? F_tgt : F_src;
  const float* GH = (c < DC) ? (sel ? G_tgt : G_src) : (sel ? H_tgt : H_src);
  int d = c & (DC - 1);
  float s = 0.f;
  for (int n = 0; n < N1C; ++n)
    s += F[((long)b * N1C + n) * DC + d] * GH[((long)b * N1C + n) * EC + e];
  float* out = sel ? Yo : Xo;
  out[(long)b * 2 * DC * EC + (long)c * EC + e] = s;
}

// coefs = block [[c1,c2],[c2,c1]], c1=0.5(l1+l1^T), c2=0.5(l2+l2^T)
__global__ void coefs_kernel(const float* __restrict__ l1, const float* __restrict__ l2,
                             float* __restrict__ co)
{
  int idx = blockIdx.x * blockDim.x + threadIdx.x;
  if (idx >= 4 * DC * DC) return;
  int c = idx / (2 * DC);
  int d = idx % (2 * DC);
  int ci = c & (DC - 1), di = d & (DC - 1);
  const float* lam = ((c >= DC) == (d >= DC)) ? l1 : l2;
  co[idx] = 0.5f * (lam[ci * DC + di] + lam[di * DC + ci]);
}

// ------------------------------------------------------------------
// Generic strided batched f32 GEMM, one wave per 16x16 C tile, using
// V_WMMA_F32_16X16X4_F32 (full fp32 precision). Grid=(M/16,N/16,B).
// A element (m,k) at A[b*aBatch + m*sam + k*sak]; B (k,n) similarly.
// C row-major MxN.
// ------------------------------------------------------------------
__global__ void __launch_bounds__(32)
wmma_gemm_f32_kernel(const float* __restrict__ A, long aBatch, int sam, int sak,
                     const float* __restrict__ Bm, long bBatch, int sbk, int sbn,
                     float* __restrict__ C, long cBatch, int N, int K)
{
  const int b = blockIdx.z;
  A  += (long)b * aBatch;
  Bm += (long)b * bBatch;
  C  += (long)b * cBatch;
  const int lane = threadIdx.x;
  const int ncol = blockIdx.y * 16 + (lane & 15);
  const int rbase = blockIdx.x * 16 + ((lane >> 4) << 3);
#if defined(__gfx1250__) && __has_builtin(__builtin_amdgcn_wmma_f32_16x16x4_f32)
  const int mrow = blockIdx.x * 16 + (lane & 15);
  const int khalf = (lane >> 4) << 1;   // lanes 16-31 hold K+2,K+3
  v8f_t acc = {0.f, 0.f, 0.f, 0.f, 0.f, 0.f, 0.f, 0.f};
  for (int k0 = 0; k0 < K; k0 += 4) {
    v2f_t a, bb;
    a.x  = A[(long)mrow * sam + (long)(k0 + khalf) * sak];
    a.y  = A[(long)mrow * sam + (long)(k0 + khalf + 1) * sak];
    bb.x = Bm[(long)(k0 + khalf) * sbk + (long)ncol * sbn];
    bb.y = Bm[(long)(k0 + khalf + 1) * sbk + (long)ncol * sbn];
    acc = __builtin_amdgcn_wmma_f32_16x16x4_f32(false, a, false, bb,
                                                (short)0, acc, false, false);
  }
#pragma unroll
  for (int r = 0; r < 8; ++r)
    C[(long)(rbase + r) * N + ncol] = acc[r];
#else
  float acc[8];
  for (int r = 0; r < 8; ++r) acc[r] = 0.f;
  for (int k = 0; k < K; ++k) {
    float bv = Bm[(long)k * sbk + (long)ncol * sbn];
    for (int r = 0; r < 8; ++r)
      acc[r] += A[(long)(rbase + r) * sam + (long)k * sak] * bv;
  }
  for (int r = 0; r < 8; ++r)
    C[(long)(rbase + r) * N + ncol] = acc[r];
#endif
}

// ------------------------------------------------------------------
// Incidence structure: find (index,value) of the single nonzero per
// (batch, edge) column of G_src/H_src/G_tgt/H_tgt.
// ------------------------------------------------------------------
__global__ void edge_idx_kernel(const float* __restrict__ Gs, const float* __restrict__ Hs,
                                const float* __restrict__ Gt, const float* __restrict__ Ht,
                                int* __restrict__ eidx, float* __restrict__ evalv)
{
  int idx = blockIdx.x * blockDim.x + threadIdx.x;
  if (idx >= 4 * BB * EC) return;
  int which = idx / (BB * EC);
  int be = idx % (BB * EC);
  int b = be / EC, e = be % EC;
  const float* M = (which == 0) ? Gs : (which == 1) ? Hs : (which == 2) ? Gt : Ht;
  int found = 0; float fv = 0.f;
  for (int n = 0; n < N1C; ++n) {
    float vv = M[((long)b * N1C + n) * EC + e];
    if (vv != 0.f) { found = n; fv = vv; break; }
  }
  eidx[idx] = found; evalv[idx] = fv;
}

__global__ void zero_f32v4_kernel(float4* __restrict__ p, long n4)
{
  long stride = (long)gridDim.x * blockDim.x;
  for (long i = (long)blockIdx.x * blockDim.x + threadIdx.x; i < n4; i += stride)
    p[i] = make_float4(0.f, 0.f, 0.f, 0.f);
}

__global__ void zero_i32_kernel(int* __restrict__ p, long n)
{
  long stride = (long)gridDim.x * blockDim.x;
  for (long i = (long)blockIdx.x * blockDim.x + threadIdx.x; i < n; i += stride)
    p[i] = 0;
}

// K[b, jq*48+ip, lq*48+kp] += Ke[b,p,q] * (incidence values). 4096 cand/batch.
__global__ void scatterK_kernel(const float* __restrict__ Ke, const int* __restrict__ eidx,
                                const float* __restrict__ evalv,
                                float* __restrict__ Kd, int* __restrict__ crow,
                                int* __restrict__ ccol)
{
  int idx = blockIdx.x * blockDim.x + threadIdx.x;
  if (idx >= BB * EC * EC) return;
  int b = idx / (EC * EC);
  int t = idx % (EC * EC);
  int p = t / EC, q = t % EC;
  const int BE = BB * EC;
  int ip = eidx[0 * BE + b * EC + p];
  int kp = eidx[1 * BE + b * EC + p];
  int jq = eidx[2 * BE + b * EC + q];
  int lq = eidx[3 * BE + b * EC + q];
  float val = Ke[(long)b * EC * EC + p * EC + q]
            * evalv[0 * BE + b * EC + p] * evalv[1 * BE + b * EC + p]
            * evalv[2 * BE + b * EC + q] * evalv[3 * BE + b * EC + q];
  int row = jq * N1C + ip;
  int col = lq * N1C + kp;
  crow[idx] = row; ccol[idx] = col;
  atomicAdd(&Kd[(long)b * NNC * NNC + (long)row * NNC + col], val);
}

// owner[t] = 1 iff no earlier candidate in same batch hits the same (row,col)
__global__ void dedup_kernel(const int* __restrict__ crow, const int* __restrict__ ccol,
                             int* __restrict__ own)
{
  int idx = blockIdx.x * blockDim.x + threadIdx.x;
  if (idx >= BB * EC * EC) return;
  int b = idx / (EC * EC);
  int t = idx % (EC * EC);
  int r = crow[idx], c = ccol[idx];
  const int* cr = crow + (long)b * EC * EC;
  const int* cc = ccol + (long)b * EC * EC;
  int o = 1;
  for (int t2 = 0; t2 < t; ++t2)
    if (cr[t2] == r && cc[t2] == c) { o = 0; break; }
  own[idx] = o;
}

// deg[b,row] = #distinct cols with K[b,row,col] > 0
__global__ void deg_kernel(const float* __restrict__ Kd, const int* __restrict__ crow,
                           const int* __restrict__ ccol, const int* __restrict__ own,
                           int* __restrict__ deg)
{
  int idx = blockIdx.x * blockDim.x + threadIdx.x;
  if (idx >= BB * EC * EC) return;
  if (!own[idx]) return;
  int b = idx / (EC * EC);
  int r = crow[idx], c = ccol[idx];
  float kv = Kd[(long)b * NNC * NNC + (long)r * NNC + c];
  if (kv > 0.f) atomicAdd(&deg[b * NNC + r], 1);
}

// ------------------------------------------------------------------
// GNN layer: x1 = mlp(emb, nf); x2 = M@x1 + mlp(emb, nsf)
// This kernel computes x1 and initializes x2 with the nsf branch.
// ------------------------------------------------------------------
__global__ void gnn_mlp_kernel(const float* __restrict__ Kp,
    const float* __restrict__ x2prev, const float* __restrict__ x6prev, int fin,
    const float* nfW0, const float* nfb0, const float* nfW1, const float* nfb1,
    const float* nsW0, const float* nsb0, const float* nsW1, const float* nsb1,
    float* __restrict__ x1out, float* __restrict__ x2out)
{
  long idx = (long)blockIdx.x * blockDim.x + threadIdx.x;
  if (idx >= (long)BB * NNC) return;
  int b = (int)(idx / NNC);
  int r = (int)(idx % NNC);
  float in[17];
  if (fin == 1) {
    // emb0[r] = Kp[b, i=r%48, j=r/48]
    in[0] = Kp[(long)b * NNC + (r % N1C) * N2C + (r / N1C)];
  } else {
#pragma unroll
    for (int f = 0; f < 16; ++f) in[f] = x2prev[idx * 16 + f];
    in[16] = x6prev[idx];
  }
  float h[16];
  // ---- nf branch ----
  if (fin == 1) {
    float x0 = in[0];
#pragma unroll
    for (int o = 0; o < 16; ++o) h[o] = fmaxf(nfb0[o] + x0 * nfW0[o], 0.f);
  } else {
    for (int o = 0; o < 16; ++o) {
      float s = nfb0[o];
#pragma unroll
      for (int i2 = 0; i2 < 17; ++i2) s += in[i2] * nfW0[i2 * 16 + o];
      h[o] = fmaxf(s, 0.f);
    }
  }
  for (int o = 0; o < 16; ++o) {
    float s = nfb1[o];
#pragma unroll
    for (int i2 = 0; i2 < 16; ++i2) s += h[i2] * nfW1[i2 * 16 + o];
    x1out[idx * 16 + o] = fmaxf(s, 0.f);
  }
  // ---- nsf branch ----
  if (fin == 1) {
    float x0 = in[0];
#pragma unroll
    for (int o = 0; o < 16; ++o) h[o] = fmaxf(nsb0[o] + x0 * nsW0[o], 0.f);
  } else {
    for (int o = 0; o < 16; ++o) {
      float s = nsb0[o];
#pragma unroll
      for (int i2 = 0; i2 < 17; ++i2) s += in[i2] * nsW0[i2 * 16 + o];
      h[o] = fmaxf(s, 0.f);
    }
  }
  for (int o = 0; o < 16; ++o) {
    float s = nsb1[o];
#pragma unroll
    for (int i2 = 0; i2 < 16; ++i2) s += h[i2] * nsW1[i2 * 16 + o];
    x2out[idx * 16 + o] = fmaxf(s, 0.f);
  }
}

// sparse M@x1 accumulation: M[r,c] = K[r,c]/deg[r] where K[r,c] > 0
__global__ void gnn_scatter_kernel(const float* __restrict__ Kd, const int* __restrict__ crow,
                                   const int* __restrict__ ccol, const int* __restrict__ own,
                                   const int* __restrict__ deg,
                                   const float* __restrict__ x1, float* __restrict__ x2)
{
  int idx = blockIdx.x * blockDim.x + threadIdx.x;
  if (idx >= BB * EC * EC) return;
  if (!own[idx]) return;
  int b = idx / (EC * EC);
  int r = crow[idx], c = ccol[idx];
  float kv = Kd[(long)b * NNC * NNC + (long)r * NNC + c];
  if (!(kv > 0.f)) return;
  float w = kv / (float)deg[b * NNC + r];
  const float* xs = x1 + ((long)b * NNC + c) * 16;
  float* xd = x2 + ((long)b * NNC + r) * 16;
#pragma unroll
  for (int f = 0; f < 16; ++f) atomicAdd(&xd[f], w * xs[f]);
}

// ------------------------------------------------------------------
// In-LDS sinkhorn (log domain), exactly mirroring the reference masks.
// sm holds raw scores s[i*48+j]; on return holds final log values.
// ------------------------------------------------------------------
static __device__ __forceinline__
void sinkhorn_lds(float* sm, int nr, int nc, int tid, int nthreads)
{
  for (int idx = tid; idx < NNC; idx += nthreads) {
    int i = idx / N2C, j = idx % N2C;
    float vv = sm[idx] / TAUC;
    bool colv = (j < nc);
    bool valid = (i < nc) && colv;
    if (i >= nr && i < nc && colv) vv = -100.0f;
    if (!valid) vv = -1.0e9f;
    sm[idx] = vv;
  }
  __syncthreads();
  for (int it = 0; it < 10; ++it) {
    if ((it & 1) == 0) {
      for (int i = tid; i < N1C; i += nthreads) {
        float m = -3.4e38f;
        for (int j = 0; j < N2C; ++j) m = fmaxf(m, sm[i * N2C + j]);
        float acc = 0.f;
        for (int j = 0; j < N2C; ++j) acc += expf(sm[i * N2C + j] - m);
        float lse = m + logf(acc);
        for (int j = 0; j < N2C; ++j) {
          bool valid = (i < nc) && (j < nc);
          sm[i * N2C + j] = valid ? (sm[i * N2C + j] - lse) : -1.0e9f;
        }
      }
    } else {
      for (int j = tid; j < N2C; j += nthreads) {
        float m = -3.4e38f;
        for (int i = 0; i < N1C; ++i) m = fmaxf(m, sm[i * N2C + j]);
        float acc = 0.f;
        for (int i = 0; i < N1C; ++i) acc += expf(sm[i * N2C + j] - m);
        float lse = m + logf(acc);
        for (int i = 0; i < N1C; ++i) {
          bool valid = (i < nc) && (j < nc);
          sm[i * N2C + j] = valid ? (sm[i * N2C + j] - lse) : -1.0e9f;
        }
      }
    }
    __syncthreads();
  }
}

// per-layer classifier + sinkhorn -> x6 feature (r = j*48+i ordering)
__global__ void gnn_cls_sinkhorn_kernel(const float* __restrict__ x2,
    const float* __restrict__ cW, const float* __restrict__ cb,
    const int* __restrict__ ns_src, const int* __restrict__ ns_tgt,
    float* __restrict__ x6out)
{
  const int b = blockIdx.x, tid = threadIdx.x;
  __shared__ float sm[NNC];
  const int nr = ns_src[b], nc = ns_tgt[b];
  const float bias = cb[0];
  for (int r = tid; r < NNC; r += blockDim.x) {
    float s = bias;
#pragma unroll
    for (int f = 0; f < 16; ++f) s += x2[((long)b * NNC + r) * 16 + f] * cW[f];
    int i = r % N1C, j = r / N1C;
    sm[i * N2C + j] = s;      // x4[i][j] = v[j*48+i]
  }
  __syncthreads();
  sinkhorn_lds(sm, nr, nc, tid, blockDim.x);
  for (int r = tid; r < NNC; r += blockDim.x) {
    int i = r % N1C, j = r / N1C;
    float vv = (i < nr && j < nc) ? expf(sm[i * N2C + j]) : 0.f;
    x6out[(long)b * NNC + r] = vv;  // x6[r=j*48+i] = sinkhorn[i][j]
  }
}

// final classifier (17 feats) + sinkhorn -> ss output [b,i,j]
__global__ void final_cls_sinkhorn_kernel(const float* __restrict__ x2,
    const float* __restrict__ x6,
    const float* __restrict__ cW, const float* __restrict__ cb,
    const int* __restrict__ ns_src, const int* __restrict__ ns_tgt,
    float* __restrict__ ssout)
{
  const int b = blockIdx.x, tid = threadIdx.x;
  __shared__ float sm[NNC];
  const int nr = ns_src[b], nc = ns_tgt[b];
  const float bias = cb[0];
  for (int r = tid; r < NNC; r += blockDim.x) {
    float s = bias;
#pragma unroll
    for (int f = 0; f < 16; ++f) s += x2[((long)b * NNC + r) * 16 + f] * cW[f];
    s += x6[(long)b * NNC + r] * cW[16];
    int i = r % N1C, j = r / N1C;
    sm[i * N2C + j] = s;
  }
  __syncthreads();
  sinkhorn_lds(sm, nr, nc, tid, blockDim.x);
  for (int idx = tid; idx < NNC; idx += blockDim.x) {
    int i = idx / N2C, j = idx % N2C;
    ssout[(long)b * NNC + idx] = (i < nr && j < nc) ? expf(sm[idx]) : 0.f;
  }
}

// ------------------------------------------------------------------
// Jonker-Volgenant LAP (max-score), one wave32 per batch. Lanes own
// columns j == lane (mod 32); min-reductions via __shfl_xor with
// first-index tie-break to match np.argmin.
// ------------------------------------------------------------------
__global__ void __launch_bounds__(32)
hungarian_kernel(const float* __restrict__ ssin, const int* __restrict__ ns_src,
                 const int* __restrict__ ns_tgt, float* __restrict__ xout)
{
  const int b = blockIdx.x;
  const int lane = threadIdx.x;
  __shared__ float cost[NNC];
  __shared__ float u[49], v[49], minv[49];
  __shared__ int way[49], p[49], usedArr[49];
  __shared__ int matchRow[N1C];
  const int n = ns_src[b];
  const int m = ns_tgt[b];
  for (int idx = lane; idx < NNC; idx += 32) cost[idx] = -ssin[(long)b * NNC + idx];
  for (int j = lane; j <= 48; j += 32) { u[j] = 0.f; v[j] = 0.f; p[j] = 0; }
  __syncthreads();
  for (int i = 1; i <= n; ++i) {
    if (lane == 0) p[0] = i;
    for (int j = lane; j <= m; j += 32) { minv[j] = 1e18f; usedArr[j] = 0; }
    __syncthreads();
    int j0 = 0;
    for (int guard = 0; guard <= m + 1; ++guard) {
      if (lane == 0) usedArr[j0] = 1;
      __syncthreads();
      const int i0 = p[j0];
      const float ui0 = u[i0];
      float bestv = 1e18f;
      int bestj = INT_MAX;
      for (int j = lane + 1; j <= m; j += 32) {
        if (!usedArr[j]) {
          float cur = cost[(i0 - 1) * N2C + (j - 1)] - ui0 - v[j];
          if (cur < minv[j]) { minv[j] = cur; way[j] = j0; }
          float mv = minv[j];
          if (mv < bestv || (mv == bestv && j < bestj)) { bestv = mv; bestj = j; }
        }
      }
      for (int off = 16; off > 0; off >>= 1) {
        float ov = __shfl_xor(bestv, off);
        int oj = __shfl_xor(bestj, off);
        if (ov < bestv || (ov == bestv && oj < bestj)) { bestv = ov; bestj = oj; }
      }
      const float delta = bestv;
      const int j1 = bestj;
      __syncthreads();
      if (j1 == INT_MAX) { j0 = 0; break; }  // safety; should not happen
      for (int j = lane; j <= m; j += 32) {
        if (usedArr[j]) { u[p[j]] += delta; v[j] -= delta; }
        else minv[j] -= delta;
      }
      __syncthreads();
      j0 = j1;
      if (p[j0] == 0) break;
    }
    if (lane == 0) {
      int jj = j0;
      while (jj != 0) { int jp = way[jj]; p[jj] = p[jp]; jj = jp; }
    }
    __syncthreads();
  }
  for (int i2 = lane; i2 < N1C; i2 += 32) matchRow[i2] = -1;
  __syncthreads();
  for (int j = lane + 1; j <= m; j += 32) { int pi = p[j]; if (pi > 0) matchRow[pi - 1] = j - 1; }
  __syncthreads();
  for (int idx = lane; idx < NNC; idx += 32) {
    int i2 = idx / N2C, j2 = idx % N2C;
    xout[(long)b * NNC + idx] = (i2 < n && matchRow[i2] == j2) ? 1.0f : 0.0f;
  }
}

// ------------------------------------------------------------------
// Host driver
// ------------------------------------------------------------------
extern "C" void kernel_launch(void* const* d_in, const int* in_sizes, int n_in,
                              void* d_out, int out_size, void* d_ws, size_t ws_size,
                              hipStream_t stream)
{
  (void)n_in; (void)out_size; (void)ws_size;
  const float* pos_src = (const float*)d_in[0];
  const float* pos_tgt = (const float*)d_in[1];
  const float* G_src = (const float*)d_in[2];
  const float* H_src = (const float*)d_in[3];
  const float* G_tgt = (const float*)d_in[4];
  const float* H_tgt = (const float*)d_in[5];
  const float* nW[3]; const float* nB[3]; const float* eW[3]; const float* eB[3];
  for (int l = 0; l < 3; ++l) {
    nW[l] = (const float*)d_in[6 + 2 * l];
    nB[l] = (const float*)d_in[7 + 2 * l];
    eW[l] = (const float*)d_in[12 + 2 * l];
    eB[l] = (const float*)d_in[13 + 2 * l];
  }
  const float* lambda1 = (const float*)d_in[18];
  const float* lambda2 = (const float*)d_in[19];
  struct Lyr { const float *nfW0,*nfb0,*nfW1,*nfb1,*nsW0,*nsb0,*nsW1,*nsb1,*cW,*cb; };
  Lyr L[3];
  for (int li = 0; li < 3; ++li) {
    int base = 20 + 10 * li;
    if (in_sizes[base + 1] == 1) {  // sorted dict keys: cls, nf, nsf
      L[li].cW   = (const float*)d_in[base + 0];
      L[li].cb   = (const float*)d_in[base + 1];
      L[li].nfW0 = (const float*)d_in[base + 2];
      L[li].nfb0 = (const float*)d_in[base + 3];
      L[li].nfW1 = (const float*)d_in[base + 4];
      L[li].nfb1 = (const float*)d_in[base + 5];
      L[li].nsW0 = (const float*)d_in[base + 6];
      L[li].nsb0 = (const float*)d_in[base + 7];
      L[li].nsW1 = (const float*)d_in[base + 8];
      L[li].nsb1 = (const float*)d_in[base + 9];
    } else {                        // insertion order: nf, nsf, cls
      L[li].nfW0 = (const float*)d_in[base + 0];
      L[li].nfb0 = (const float*)d_in[base + 1];
      L[li].nfW1 = (const float*)d_in[base + 2];
      L[li].nfb1 = (const float*)d_in[base + 3];
      L[li].nsW0 = (const float*)d_in[base + 4];
      L[li].nsb0 = (const float*)d_in[base + 5];
      L[li].nsW1 = (const float*)d_in[base + 6];
      L[li].nsb1 = (const float*)d_in[base + 7];
      L[li].cW   = (const float*)d_in[base + 8];
      L[li].cb   = (const float*)d_in[base + 9];
    }
  }
  const float* cls_W = (const float*)d_in[50];
  const float* cls_b = (const float*)d_in[51];
  const int* ns_src = (const int*)d_in[52];
  const int* ns_tgt = (const int*)d_in[53];

  float* wsF = (float*)d_ws;
  int* wsI = (int*)(wsF + WS_FEND);
  float* outF = (float*)d_out;

  // 1) node/edge MLPs -> U_src, F_src, U_tgt, F_tgt
  mlp_kernel<<<dim3((4 * BB * N1C + 255) / 256), dim3(256), 0, stream>>>(
      pos_src, pos_tgt,
      nW[0], nB[0], nW[1], nB[1], nW[2], nB[2],
      eW[0], eB[0], eW[1], eB[1], eW[2], eB[2],
      wsF + WS_U_SRC, wsF + WS_F_SRC, wsF + WS_U_TGT, wsF + WS_F_TGT);

  // 2) X, Y edge-feature aggregations
  long xyTot = 2L * BB * 2 * DC * EC;
  xy_kernel<<<dim3((unsigned)((xyTot + 255) / 256)), dim3(256), 0, stream>>>(
      wsF + WS_F_SRC, G_src, H_src, wsF + WS_F_TGT, G_tgt, H_tgt,
      wsF + WS_X, wsF + WS_Y);

  // 3) coefs block matrix
  coefs_kernel<<<dim3(256), dim3(256), 0, stream>>>(lambda1, lambda2, wsF + WS_CO);

  // 4) WMMA f32 GEMMs: Z = coefs @ Y[b]; Ke = X[b]^T @ Z[b]; Kp = U_src @ U_tgt^T
  wmma_gemm_f32_kernel<<<dim3(16, 4, BB), dim3(32), 0, stream>>>(
      wsF + WS_CO, 0L, 2 * DC, 1,
      wsF + WS_Y, (long)2 * DC * EC, EC, 1,
      wsF + WS_Z, (long)2 * DC * EC, EC, 2 * DC);
  wmma_gemm_f32_kernel<<<dim3(4, 4, BB), dim3(32), 0, stream>>>(
      wsF + WS_X, (long)2 * DC * EC, 1, EC,
      wsF + WS_Z, (long)2 * DC * EC, EC, 1,
      wsF + WS_KE, (long)EC * EC, EC, 2 * DC);
  wmma_gemm_f32_kernel<<<dim3(3, 3, BB), dim3(32), 0, stream>>>(
      wsF + WS_U_SRC, (long)N1C * DC, DC, 1,
      wsF + WS_U_TGT, (long)N2C * DC, 1, DC,
      wsF + WS_KP, (long)N1C * N2C, N2C, DC);

  // 5) incidence structure + sparse K scatter (K output region)
  edge_idx_kernel<<<dim3(16), dim3(256), 0, stream>>>(
      G_src, H_src, G_tgt, H_tgt, wsI + WI_EIDX, wsF + WS_EVAL);
  zero_f32v4_kernel<<<dim3(4096), dim3(256), 0, stream>>>(
      (float4*)(outF + K_OFF), K_ELEMS / 4);
  scatterK_kernel<<<dim3(256), dim3(256), 0, stream>>>(
      wsF + WS_KE, wsI + WI_EIDX, wsF + WS_EVAL,
      outF + K_OFF, wsI + WI_CROW, wsI + WI_CCOL);
  dedup_kernel<<<dim3(256), dim3(256), 0, stream>>>(
      wsI + WI_CROW, wsI + WI_CCOL, wsI + WI_OWN);
  zero_i32_kernel<<<dim3(144), dim3(256), 0, stream>>>(wsI + WI_DEG, (long)BB * NNC);
  deg_kernel<<<dim3(256), dim3(256), 0, stream>>>(
      outF + K_OFF, wsI + WI_CROW, wsI + WI_CCOL, wsI + WI_OWN, wsI + WI_DEG);

  // 6) GNN layers (sparse message passing + per-layer sinkhorn feature)
  float* x2buf[2] = { wsF + WS_X2A, wsF + WS_X2B };
  float* x6buf[2] = { wsF + WS_X6A, wsF + WS_X6B };
  int cur = 0;
  const float* x2prev = nullptr;
  const float* x6prev = nullptr;
  for (int li = 0; li < 3; ++li) {
    float* x2c = x2buf[cur];
    float* x6c = x6buf[cur];
    int fin = (li == 0) ? 1 : 17;
    gnn_mlp_kernel<<<dim3(144), dim3(256), 0, stream>>>(
        wsF + WS_KP, x2prev, x6prev, fin,
        L[li].nfW0, L[li].nfb0, L[li].nfW1, L[li].nfb1,
        L[li].nsW0, L[li].nsb0, L[li].nsW1, L[li].nsb1,
        wsF + WS_X1, x2c);
    gnn_scatter_kernel<<<dim3(256), dim3(256), 0, stream>>>(
        outF + K_OFF, wsI + WI_CROW, wsI + WI_CCOL, wsI + WI_OWN, wsI + WI_DEG,
        wsF + WS_X1, x2c);
    gnn_cls_sinkhorn_kernel<<<dim3(BB), dim3(64), 0, stream>>>(
        x2c, L[li].cW, L[li].cb, ns_src, ns_tgt, x6c);
    x2prev = x2c; x6prev = x6c; cur ^= 1;
  }

  // 7) final classifier + sinkhorn -> ss
  final_cls_sinkhorn_kernel<<<dim3(BB), dim3(64), 0, stream>>>(
      x2prev, x6prev, cls_W, cls_b, ns_src, ns_tgt, outF + SS_OFF);

  // 8) Hungarian matching -> x
  hungarian_kernel<<<dim3(BB), dim3(32), 0, stream>>>(
      outF + SS_OFF, ns_src, ns_tgt, outF + X_OFF);
}